// PTAE_70781061038699
// MI455X (gfx1250) — compile-verified
//
#include <hip/hip_runtime.h>
#include <math.h>

#define T_SEQ   512
#define BATCH   256
#define INPUT   512
#define HIDDEN  512
#define GATES   2048            // 4*HIDDEN
#define KTOT    1024            // INPUT+HIDDEN
#define BH      (BATCH*HIDDEN)  // 131072
#define FORGET_BIAS 0.8f
#define NBLK    32

typedef __attribute__((ext_vector_type(2))) float v2f;
typedef __attribute__((ext_vector_type(8))) float v8f;

__device__ __forceinline__ v8f wmma4(v2f a, v2f b, v8f c) {
    // D = A(16x4 f32) x B(4x16 f32) + C(16x16 f32)
    return __builtin_amdgcn_wmma_f32_16x16x4_f32(
        /*neg_a=*/false, a, /*neg_b=*/false, b,
        /*c_mod=*/(short)0, c, /*reuse_a=*/false, /*reuse_b=*/false);
}

__device__ __forceinline__ float sigm(float x) { return 1.0f / (1.0f + __expf(-x)); }

// One-time LDS-tiled transpose: Wt[n][k] = W[k][n]
// W: (KTOT=1024, GATES=2048) -> Wt: (GATES=2048, KTOT=1024)
// grid: (GATES/32, KTOT/32), block: 256 = 32x8
__global__ __launch_bounds__(256) void transpose_w(const float* __restrict__ W,
                                                   float* __restrict__ Wt) {
    __shared__ float tile[32][33];          // +1 pad kills bank conflicts
    const int tx = threadIdx.x & 31;
    const int ty = threadIdx.x >> 5;        // 0..7
    const int n0 = blockIdx.x * 32;
    const int k0 = blockIdx.y * 32;
    #pragma unroll
    for (int i = 0; i < 32; i += 8)         // coalesced read of W rows
        tile[ty + i][tx] = W[(size_t)(k0 + ty + i) * GATES + (n0 + tx)];
    __syncthreads();
    #pragma unroll
    for (int i = 0; i < 32; i += 8)         // coalesced write of Wt rows
        Wt[(size_t)(n0 + ty + i) * KTOT + (k0 + tx)] = tile[tx][ty + i];
}

__global__ __launch_bounds__(256)
void lstm_persistent(const float* __restrict__ x,      // (T,B,INPUT)
                     const float* __restrict__ Wt,     // (GATES,KTOT) transposed
                     const float* __restrict__ bias,   // (GATES)
                     const float* __restrict__ peep_i,
                     const float* __restrict__ peep_f,
                     const float* __restrict__ peep_o,
                     float*       __restrict__ cbuf,   // (B,H) state
                     unsigned*    __restrict__ bar,    // arrive counter
                     float*       __restrict__ out)    // hidden_seq | h_T | c_T
{
    const int lane = threadIdx.x & 31;
    const int wave = threadIdx.x >> 5;
    const int lh   = lane >> 4;    // lane-half selects K pair / M+8
    const int ln   = lane & 15;    // M (A frag) or N (B/C frag)
    const int blockM = blockIdx.x & 7;
    const int blockN = blockIdx.x >> 3;
    const int m0 = blockM * 32;                    // batch-row tile (32 rows/block)
    const int n  = blockN * 128 + wave * 16 + ln;  // hidden column owned by lane

    // Per-lane constants for gate math (column n of each gate).
    const float bi = bias[n];
    const float bj = bias[HIDDEN + n];
    const float bf = bias[2 * HIDDEN + n] + FORGET_BIAS;
    const float bo = bias[3 * HIDDEN + n];
    const float pi = peep_i[n];
    const float pf = peep_f[n];
    const float po = peep_o[n];

    // B-operand row pointers (K-contiguous thanks to transpose), one per gate.
    const float* wt0 = Wt + (size_t)(0 * HIDDEN + n) * KTOT + 2 * lh;
    const float* wt1 = Wt + (size_t)(1 * HIDDEN + n) * KTOT + 2 * lh;
    const float* wt2 = Wt + (size_t)(2 * HIDDEN + n) * KTOT + 2 * lh;
    const float* wt3 = Wt + (size_t)(3 * HIDDEN + n) * KTOT + 2 * lh;

    for (int t = 0; t < T_SEQ; ++t) {
        v8f aI0 = {}, aI1 = {}, aJ0 = {}, aJ1 = {};
        v8f aF0 = {}, aF1 = {}, aO0 = {}, aO1 = {};

        // ---------- K part 1: x_t @ W[:INPUT]  (no dependence on h_{t-1}) ----------
        // x is streamed exactly once (256 MB > 192 MB L2): mark non-temporal so
        // it does not evict the L2-resident Wt / h / c working set.
        const float* xr0 = x + ((size_t)t * BATCH + m0 + ln) * INPUT + 2 * lh;
        const float* xr1 = xr0 + 16 * INPUT;   // second M tile (+16 rows)
        #pragma unroll 4
        for (int k = 0; k < INPUT; k += 4) {
            v2f a0 = __builtin_nontemporal_load((const v2f*)(xr0 + k));
            v2f a1 = __builtin_nontemporal_load((const v2f*)(xr1 + k));
            v2f b0 = *(const v2f*)(wt0 + k);
            v2f b1 = *(const v2f*)(wt1 + k);
            v2f b2 = *(const v2f*)(wt2 + k);
            v2f b3 = *(const v2f*)(wt3 + k);
            aI0 = wmma4(a0, b0, aI0);  aI1 = wmma4(a1, b0, aI1);
            aJ0 = wmma4(a0, b1, aJ0);  aJ1 = wmma4(a1, b1, aJ1);
            aF0 = wmma4(a0, b2, aF0);  aF1 = wmma4(a1, b2, aF1);
            aO0 = wmma4(a0, b3, aO0);  aO1 = wmma4(a1, b3, aO1);
        }

        // ---------- device-wide wait: h_{t-1} fully published ----------
        if (t > 0) {
            if (threadIdx.x == 0) {
                const unsigned target = (unsigned)NBLK * (unsigned)t;
                while (__hip_atomic_load(bar, __ATOMIC_RELAXED,
                                         __HIP_MEMORY_SCOPE_AGENT) < target)
                    __builtin_amdgcn_s_sleep(1);
            }
            __syncthreads();
            __threadfence();  // acquire

            // ---------- K part 2: h_{t-1} @ W[INPUT:] ----------
            const float* hr0 = out + (size_t)(t - 1) * BH
                                   + (size_t)(m0 + ln) * HIDDEN + 2 * lh;
            const float* hr1 = hr0 + 16 * HIDDEN;
            #pragma unroll 4
            for (int k = 0; k < HIDDEN; k += 4) {
                v2f a0 = *(const v2f*)(hr0 + k);
                v2f a1 = *(const v2f*)(hr1 + k);
                v2f b0 = *(const v2f*)(wt0 + INPUT + k);
                v2f b1 = *(const v2f*)(wt1 + INPUT + k);
                v2f b2 = *(const v2f*)(wt2 + INPUT + k);
                v2f b3 = *(const v2f*)(wt3 + INPUT + k);
                aI0 = wmma4(a0, b0, aI0);  aI1 = wmma4(a1, b0, aI1);
                aJ0 = wmma4(a0, b1, aJ0);  aJ1 = wmma4(a1, b1, aJ1);
                aF0 = wmma4(a0, b2, aF0);  aF1 = wmma4(a1, b2, aF1);
                aO0 = wmma4(a0, b3, aO0);  aO1 = wmma4(a1, b3, aO1);
            }
        }

        // ---------- gates + state update (all four gates live in this lane) ----------
        float* houtT = out + (size_t)t * BH;
        #pragma unroll
        for (int mt = 0; mt < 2; ++mt) {
            v8f gi = mt ? aI1 : aI0;
            v8f gj = mt ? aJ1 : aJ0;
            v8f gf = mt ? aF1 : aF0;
            v8f go = mt ? aO1 : aO0;
            const int rbase = m0 + 16 * mt + 8 * lh;  // C layout: M = v (+8 hi half)
            #pragma unroll
            for (int v = 0; v < 8; ++v) {
                const size_t off = (size_t)(rbase + v) * HIDDEN + n;
                float cp = cbuf[off];
                float iv = gi[v] + bi;
                float jv = gj[v] + bj;
                float fv = gf[v] + bf;     // includes FORGET_BIAS
                float ov = go[v] + bo;
                float cn = sigm(fv + cp * pf) * cp + sigm(iv + cp * pi) * tanhf(jv);
                float hn = sigm(ov + cn * po) * tanhf(cn);
                cbuf[off]  = cn;
                houtT[off] = hn;
                if (t == T_SEQ - 1) {
                    out[(size_t)T_SEQ * BH + off]      = hn;  // h_T
                    out[(size_t)T_SEQ * BH + BH + off] = cn;  // c_T
                }
            }
        }

        // ---------- arrive ----------
        __syncthreads();
        __threadfence();  // release h_t / c
        if (threadIdx.x == 0)
            __hip_atomic_fetch_add(bar, 1u, __ATOMIC_RELEASE,
                                   __HIP_MEMORY_SCOPE_AGENT);
    }
}

extern "C" void kernel_launch(void* const* d_in, const int* in_sizes, int n_in,
                              void* d_out, int out_size, void* d_ws, size_t ws_size,
                              hipStream_t stream) {
    const float* x      = (const float*)d_in[0];
    const float* W      = (const float*)d_in[1];
    const float* bias   = (const float*)d_in[2];
    const float* peep_i = (const float*)d_in[3];
    const float* peep_f = (const float*)d_in[4];
    const float* peep_o = (const float*)d_in[5];
    float* out = (float*)d_out;

    // Scratch layout: [cbuf: B*H f32][Wt: 2048x1024 f32][barrier: 64B]
    float*    cbuf = (float*)d_ws;
    float*    Wt   = (float*)((char*)d_ws + (size_t)BH * 4);
    unsigned* bar  = (unsigned*)((char*)d_ws + (size_t)BH * 4 + (size_t)GATES * KTOT * 4);

    hipMemsetAsync(cbuf, 0, (size_t)BH * 4, stream);   // c_0 = 0
    hipMemsetAsync(bar, 0, 64, stream);                // reset arrive counter
    dim3 tgrid(GATES / 32, KTOT / 32);
    transpose_w<<<tgrid, 256, 0, stream>>>(W, Wt);
    lstm_persistent<<<NBLK, 256, 0, stream>>>(x, Wt, bias, peep_i, peep_f, peep_o,
                                              cbuf, bar, out);
}